// TE_loss_21981642621252
// MI455X (gfx1250) — compile-verified
//
#include <hip/hip_runtime.h>
#include <hip/hip_bf16.h>
#include <stdint.h>

// Problem constants (from reference)
#define BB 8
#define SS 160
#define VV 32000
#define MM 32
#define SGEN 64
#define EMAX 8

typedef float v2f __attribute__((ext_vector_type(2)));
typedef float v8f __attribute__((ext_vector_type(8)));

// ---------------------------------------------------------------------------
// CDNA5 async global->LDS copy (ASYNCcnt path), inline asm per bridge doc.
// LDS address is taken from the VDST VGPR; flat shared address low 32 bits == LDS offset.
// ---------------------------------------------------------------------------
__device__ __forceinline__ void async_copy_b128(uint32_t lds_off, const float* g) {
    asm volatile("global_load_async_to_lds_b128 %0, %1, off"
                 :: "v"(lds_off), "v"((uint64_t)(uintptr_t)g)
                 : "memory");
}
__device__ __forceinline__ void wait_async0() {
    asm volatile("s_wait_asynccnt 0x0" ::: "memory");
}

// ---------------------------------------------------------------------------
// One-pass (online) logsumexp over a row of VV floats, block = 256 threads.
// Double-buffered async tiles: 1024 floats (4KB) per tile, 32 tiles.
// Fetch of tile t+1 overlaps compute of tile t.
// ---------------------------------------------------------------------------
__device__ float row_logsumexp(const float* __restrict__ row) {
    __shared__ __align__(16) float tiles[2][1024];
    __shared__ float redm[256];
    __shared__ float reds[256];

    const int tid = threadIdx.x;
    const int ntiles = (VV + 1023) >> 10;   // 32 (last tile: 256 floats)

    float mx = -3.402823466e38f;
    float sm = 0.0f;

    // prologue: tile 0
    {
        const int idx = tid * 4;
        if (idx < VV)
            async_copy_b128((uint32_t)(uintptr_t)&tiles[0][tid * 4], row + idx);
    }
    wait_async0();
    __syncthreads();

    for (int t = 0; t < ntiles; ++t) {
        // issue fetch of tile t+1 into the other buffer (overlaps compute below)
        if (t + 1 < ntiles) {
            const int nidx = (t + 1) * 1024 + tid * 4;
            if (nidx < VV)
                async_copy_b128((uint32_t)(uintptr_t)&tiles[(t + 1) & 1][tid * 4],
                                row + nidx);
        }
        // consume tile t from LDS
        const int idx = t * 1024 + tid * 4;
        if (idx < VV) {
            const float4 v = *reinterpret_cast<const float4*>(&tiles[t & 1][tid * 4]);
            const float xs[4] = {v.x, v.y, v.z, v.w};
#pragma unroll
            for (int e = 0; e < 4; ++e) {
                const float x  = xs[e];
                const float nm = fmaxf(mx, x);
                sm = sm * __expf(mx - nm) + __expf(x - nm);
                mx = nm;
            }
        }
        if (t + 1 < ntiles) wait_async0();   // own prefetch landed in LDS
        __syncthreads();                     // cross-wave: data visible, buffer reusable
    }

    // block reduction of (max, scaled-sum) pairs
    redm[tid] = mx;
    reds[tid] = sm;
    __syncthreads();
    for (int off = 128; off >= 1; off >>= 1) {
        if (tid < off) {
            const float m1 = redm[tid],       s1 = reds[tid];
            const float m2 = redm[tid + off], s2 = reds[tid + off];
            const float Mx = fmaxf(m1, m2);
            reds[tid] = s1 * __expf(m1 - Mx) + s2 * __expf(m2 - Mx);
            redm[tid] = Mx;
        }
        __syncthreads();
    }
    return redm[0] + __logf(reds[0]);
}

// ---------------------------------------------------------------------------
// Kernel A: LSE over all 2048 gen_cap_preds rows. block = (m*64+s).
// ---------------------------------------------------------------------------
__global__ void __launch_bounds__(256)
lse_gen_kernel(const float* __restrict__ gen, float* __restrict__ lse_out) {
    const int rs = blockIdx.x;                     // m*SGEN + s
    const float lse = row_logsumexp(gen + (size_t)rs * VV);
    if (threadIdx.x == 0) lse_out[rs] = lse;
}

// ---------------------------------------------------------------------------
// Kernel B: G[m*8+j] = sum_s gen[m,s,entities[m,j]] via V_WMMA_F32_16X16X4_F32
// with all-ones B. One wave handles 16 (m,j) rows; K chunks of 4 over s.
// A layout (ISA 16x4 f32): lanes 0-15 -> M=lane, K={0,1}; lanes 16-31 -> K={2,3}.
// D layout: lane<16 elem i => (M=i, N=lane); lane>=16 elem i => (M=8+i).
// ---------------------------------------------------------------------------
__global__ void __launch_bounds__(32)
gather_sum_wmma_kernel(const float* __restrict__ gen,
                       const int* __restrict__ entities,
                       float* __restrict__ G) {
    const int base = blockIdx.x * 16;              // rows base .. base+15 (row = m*8+j)
    const int lane = threadIdx.x;                  // 0..31, EXEC all ones (no divergence)
    const int lrow = base + (lane & 15);
    const int m    = lrow >> 3;
    const int e    = entities[lrow];
    const float* p = gen + (size_t)m * SGEN * VV + e;   // stride VV over s
    const int soff = (lane >> 4) * 2;              // K sub-offset per ISA layout

    v8f c = {0.f, 0.f, 0.f, 0.f, 0.f, 0.f, 0.f, 0.f};
    v2f b;  b.x = 1.0f;  b.y = 1.0f;               // ones => D[r][n] = sum_k A[r][k] + C

#pragma unroll 4
    for (int kc = 0; kc < SGEN / 4; ++kc) {
        const int s0 = kc * 4 + soff;
        v2f a;
        a.x = p[(size_t)s0 * VV];
        a.y = p[(size_t)(s0 + 1) * VV];
        c = __builtin_amdgcn_wmma_f32_16x16x4_f32(false, a, false, b,
                                                  (short)0, c, false, false);
    }

    if (lane == 0 || lane == 16) {                 // N=0 columns hold the row sums
        const int r0 = base + ((lane == 16) ? 8 : 0);
#pragma unroll
        for (int i = 0; i < 8; ++i) G[r0 + i] = c[i];
    }
}

// ---------------------------------------------------------------------------
// Kernel C: LSE + gathered logit for the 256 needed preds rows. block=(m*8+j).
// ---------------------------------------------------------------------------
__global__ void __launch_bounds__(256)
lse_pred_kernel(const float* __restrict__ preds,
                const int* __restrict__ sample_index,
                const int* __restrict__ entities,
                const int* __restrict__ prefix_lens,
                const int* __restrict__ prompt_length_p,
                float* __restrict__ lse_out,
                float* __restrict__ logit_out) {
    const int row = blockIdx.x;                    // m*8 + j
    const int m = row >> 3, j = row & 7;
    const int pl  = prompt_length_p[0];
    int pos = pl + prefix_lens[m] - 1 + j;
    pos = pos < 0 ? 0 : (pos > SS - 1 ? SS - 1 : pos);
    const float* rp = preds + ((size_t)sample_index[m] * SS + pos) * VV;
    const float lse = row_logsumexp(rp);
    if (threadIdx.x == 0) {
        lse_out[row]   = lse;
        logit_out[row] = rp[entities[row]];
    }
}

// ---------------------------------------------------------------------------
// Kernel D: finalize. One wave32, lane m = sample m (M == 32 == wave size).
// ---------------------------------------------------------------------------
__global__ void __launch_bounds__(32)
finalize_kernel(const int* __restrict__ entity_counts,
                const int* __restrict__ prefix_lens,
                const int* __restrict__ prompt_length_p,
                const float* __restrict__ lse_gen,
                const float* __restrict__ G,
                const float* __restrict__ lse_pred,
                const float* __restrict__ logit,
                float* __restrict__ out) {
    const int m  = threadIdx.x;                    // 0..31
    const int pl = prompt_length_p[0];

    float ml = 0.0f;                               // mean LSE over gen steps
    for (int s = 0; s < SGEN; ++s) ml += lse_gen[m * SGEN + s];
    ml *= (1.0f / SGEN);

    const int cnt = entity_counts[m];
    const int pf  = prefix_lens[m];
    float sl = 0.0f;
#pragma unroll
    for (int j = 0; j < EMAX; ++j) {
        const int pos = pl + pf - 1 + j;
        const bool valid = (j < cnt) && (pos < SS);
        const int r = m * EMAX + j;
        const float pred_kw = logit[r] - lse_pred[r];
        const float gen_kw  = G[r] * (1.0f / SGEN) - ml;
        sl += valid ? (pred_kw - gen_kw) : 0.0f;
    }
    float v = sl / (float)cnt;
    for (int off = 16; off >= 1; off >>= 1) v += __shfl_xor(v, off);
    if (m == 0) out[0] = -v / (float)MM;
}

// ---------------------------------------------------------------------------
extern "C" void kernel_launch(void* const* d_in, const int* in_sizes, int n_in,
                              void* d_out, int out_size, void* d_ws, size_t ws_size,
                              hipStream_t stream) {
    (void)in_sizes; (void)n_in; (void)out_size; (void)ws_size;
    const float* preds          = (const float*)d_in[0];
    const float* gen            = (const float*)d_in[1];
    const int*   sample_index   = (const int*)d_in[2];
    const int*   entities       = (const int*)d_in[3];
    const int*   entity_counts  = (const int*)d_in[4];
    const int*   prefix_lens    = (const int*)d_in[5];
    const int*   prompt_length  = (const int*)d_in[6];

    float* ws       = (float*)d_ws;
    float* lse_gen  = ws;            // 2048 floats
    float* G        = ws + 2048;     // 256 floats
    float* lse_pred = ws + 2304;     // 256 floats
    float* logit    = ws + 2560;     // 256 floats

    hipLaunchKernelGGL(lse_gen_kernel, dim3(MM * SGEN), dim3(256), 0, stream,
                       gen, lse_gen);
    hipLaunchKernelGGL(gather_sum_wmma_kernel, dim3((MM * EMAX) / 16), dim3(32), 0, stream,
                       gen, entities, G);
    hipLaunchKernelGGL(lse_pred_kernel, dim3(MM * EMAX), dim3(256), 0, stream,
                       preds, sample_index, entities, prefix_lens, prompt_length,
                       lse_pred, logit);
    hipLaunchKernelGGL(finalize_kernel, dim3(1), dim3(32), 0, stream,
                       entity_counts, prefix_lens, prompt_length,
                       lse_gen, G, lse_pred, logit, (float*)d_out);
}